// PCN_encoder_emb_25744033972952
// MI455X (gfx1250) — compile-verified
//
#include <hip/hip_runtime.h>
#include <stdint.h>

typedef __attribute__((ext_vector_type(2))) float v2f;
typedef __attribute__((ext_vector_type(8))) float v8f;

#define NPTS 8192
#define YS 136   // LDS row stride (floats) for Y tile: 2*136 % 64 == 16 -> conflict-free dual-row reads
#define HS 136   // LDS row stride for H tile

// ---- ordered-float encoding for atomic max ----
__device__ __forceinline__ unsigned ordf(float f) {
  unsigned u = __float_as_uint(f);
  return (u & 0x80000000u) ? ~u : (u | 0x80000000u);
}
__device__ __forceinline__ float unordf(unsigned e) {
  return __uint_as_float((e & 0x80000000u) ? (e ^ 0x80000000u) : ~e);
}

// ---- zero-init workspace region used by atomics ----
__global__ void init_zero_kernel(unsigned* p, int n) {
  int i = blockIdx.x * blockDim.x + threadIdx.x;
  if (i < n) p[i] = 0u;
}

// ---- furthest point sampling: one workgroup per batch, xyz+dist in LDS ----
__global__ __launch_bounds__(1024) void fps_kernel(const float* __restrict__ x,
                                                   int* __restrict__ idx_out,
                                                   int npoint) {
  __shared__ float px[NPTS], py[NPTS], pz[NPTS], dist[NPTS];
  __shared__ float wv[32];
  __shared__ int wi[32];
  __shared__ int s_last;
  const int b = blockIdx.x;
  const int t = threadIdx.x;
  const float* xb = x + (size_t)b * 3 * NPTS;
  for (int n = t; n < NPTS; n += 1024) {
    px[n] = xb[n];
    py[n] = xb[NPTS + n];
    pz[n] = xb[2 * NPTS + n];
    dist[n] = 1e10f;
  }
  if (t == 0) { idx_out[(size_t)b * npoint] = 0; s_last = 0; }
  __syncthreads();
  const int wave = t >> 5, lane = t & 31;
  for (int s = 1; s < npoint; ++s) {
    const int last = s_last;
    const float lx = px[last], ly = py[last], lz = pz[last];
    float bv = -1.0f; int bi = 0;
    for (int n = t; n < NPTS; n += 1024) {
      float dx = px[n] - lx, dy = py[n] - ly, dz = pz[n] - lz;
      float d = dx * dx + dy * dy + dz * dz;
      float dm = fminf(dist[n], d);
      dist[n] = dm;
      if (dm > bv) { bv = dm; bi = n; }   // strict > keeps lowest index (argmax-first semantics)
    }
    for (int off = 16; off > 0; off >>= 1) {
      float ov = __shfl_xor(bv, off, 32);
      int oi = __shfl_xor(bi, off, 32);
      if (ov > bv || (ov == bv && oi < bi)) { bv = ov; bi = oi; }
    }
    if (lane == 0) { wv[wave] = bv; wi[wave] = bi; }
    __syncthreads();
    if (wave == 0) {
      float v = wv[lane]; int i2 = wi[lane];
      for (int off = 16; off > 0; off >>= 1) {
        float ov = __shfl_xor(v, off, 32);
        int oi = __shfl_xor(i2, off, 32);
        if (ov > v || (ov == v && oi < i2)) { v = ov; i2 = oi; }
      }
      if (lane == 0) { s_last = i2; idx_out[(size_t)b * npoint + s] = i2; }
    }
    __syncthreads();
  }
}

// ---- 3->16 (relu) -> 64 conv + per-batch channel max (branch gf vectors) ----
__global__ __launch_bounds__(256) void branch_conv_kernel(
    const float* __restrict__ x, const int* __restrict__ idx, int npts,
    const float* __restrict__ w1, const float* __restrict__ b1,
    const float* __restrict__ w2, const float* __restrict__ b2,
    unsigned* __restrict__ ordmax) {
  __shared__ float sw1[48], sb1[16], sw2[1024], sb2[64];
  __shared__ unsigned smax[64];
  const int b = blockIdx.y;
  const int t = threadIdx.x;
  for (int j = t; j < 48; j += 256) sw1[j] = w1[j];
  for (int j = t; j < 16; j += 256) sb1[j] = b1[j];
  for (int j = t; j < 1024; j += 256) sw2[j] = w2[j];
  for (int j = t; j < 64; j += 256) sb2[j] = b2[j];
  if (t < 64) smax[t] = 0u;
  __syncthreads();
  const int i = blockIdx.x * 256 + t;
  if (i < npts) {
    const int n = idx ? idx[(size_t)b * npts + i] : i;
    const float* xb = x + (size_t)b * 3 * NPTS;
    const float pxv = xb[n], pyv = xb[NPTS + n], pzv = xb[2 * NPTS + n];
    float h[16];
#pragma unroll
    for (int c = 0; c < 16; c++) {
      float v = sw1[c * 3] * pxv + sw1[c * 3 + 1] * pyv + sw1[c * 3 + 2] * pzv + sb1[c];
      h[c] = fmaxf(v, 0.0f);
    }
    for (int o = 0; o < 64; o++) {
      float acc = sb2[o];
#pragma unroll
      for (int c = 0; c < 16; c++) acc += sw2[o * 16 + c] * h[c];
      atomicMax(&smax[o], ordf(acc));
    }
  }
  __syncthreads();
  if (t < 64) atomicMax(&ordmax[(size_t)b * 64 + t], smax[t]);
}

// ---- per-batch constant: cb[b][m] = b4[m] + w4[m,64:256] . [gf1;gf2;gf3] ----
__global__ __launch_bounds__(256) void cb_kernel(
    const unsigned* __restrict__ ord0, const unsigned* __restrict__ ord1,
    const unsigned* __restrict__ ord2,
    const float* __restrict__ w4, const float* __restrict__ b4,
    float* __restrict__ cbv) {
  __shared__ float gf[192];
  const int b = blockIdx.x;
  const int m = threadIdx.x;
  if (m < 64) gf[m] = unordf(ord0[b * 64 + m]);
  else if (m < 128) gf[m] = unordf(ord1[b * 64 + (m - 64)]);
  else if (m < 192) gf[m] = unordf(ord2[b * 64 + (m - 128)]);
  __syncthreads();
  float acc = b4[m];
  const float* wr = w4 + (size_t)m * 256 + 64;
  for (int j = 0; j < 192; j++) acc += wr[j] * gf[j];
  cbv[b * 256 + m] = acc;
}

// ---- fused main kernel: recompute y tile -> GEMM1 (WMMA) -> relu -> GEMM2 (WMMA) ----
__global__ __launch_bounds__(256) void main_gemm_kernel(
    const float* __restrict__ x,
    const float* __restrict__ w1, const float* __restrict__ b1,
    const float* __restrict__ w2, const float* __restrict__ b2,
    const float* __restrict__ w4, const float* __restrict__ cbv,
    const float* __restrict__ w6, const float* __restrict__ b6,
    float* __restrict__ out) {
  __shared__ float ylds[64 * YS];    // Y tile: 64 x 128
  __shared__ float hlds[256 * HS];   // H tile: 256 x 128
  const int b = blockIdx.y;
  const int n0 = blockIdx.x * 128;
  const int t = threadIdx.x;

  // stage 0: small 3->16->64 conv for this 128-point tile, directly into LDS
  {
    const int p = t & 127;
    const int halfc = (t >> 7) * 32;
    const int n = n0 + p;
    const float* xb = x + (size_t)b * 3 * NPTS;
    const float pxv = xb[n], pyv = xb[NPTS + n], pzv = xb[2 * NPTS + n];
    float h16[16];
#pragma unroll
    for (int c = 0; c < 16; c++) {
      float v = w1[c * 3] * pxv + w1[c * 3 + 1] * pyv + w1[c * 3 + 2] * pzv + b1[c];
      h16[c] = fmaxf(v, 0.0f);
    }
    for (int oo = 0; oo < 32; oo++) {
      const int o = halfc + oo;
      float acc = b2[o];
#pragma unroll
      for (int c = 0; c < 16; c++) acc += w2[o * 16 + c] * h16[c];
      ylds[o * YS + p] = acc;
    }
  }
  __syncthreads();

  const int wave = t >> 5, lane = t & 31;
  const int ln = lane & 15, hl = lane >> 4;
  const int colbase = wave * 16 + ln;   // each wave owns a fixed 16-column slab

  // GEMM1: H[256 x 128] = relu( W4[:, :64] * Y + cb )
  for (int mg = 0; mg < 16; mg += 4) {
    v8f acc[4] = {};
    for (int k = 0; k < 64; k += 4) {
      const int kk = k + 2 * hl;
      v2f bf;
      bf.x = ylds[kk * YS + colbase];
      bf.y = ylds[(kk + 1) * YS + colbase];
#pragma unroll
      for (int mi = 0; mi < 4; mi++) {
        const float* wr = w4 + (size_t)((mg + mi) * 16 + ln) * 256 + kk;
        v2f af; af.x = wr[0]; af.y = wr[1];
        acc[mi] = __builtin_amdgcn_wmma_f32_16x16x4_f32(
            false, af, false, bf, (short)0, acc[mi], false, false);
      }
    }
#pragma unroll
    for (int mi = 0; mi < 4; mi++) {
#pragma unroll
      for (int i = 0; i < 8; i++) {
        const int m = (mg + mi) * 16 + i + 8 * hl;
        float v = acc[mi][i] + cbv[b * 256 + m];
        hlds[m * HS + colbase] = fmaxf(v, 0.0f);
      }
    }
  }
  __syncthreads();

  // GEMM2: OUT[256 x 128] = W6 * H + b6
  for (int mg = 0; mg < 16; mg += 4) {
    v8f acc[4] = {};
    for (int k = 0; k < 256; k += 4) {
      const int kk = k + 2 * hl;
      v2f bf;
      bf.x = hlds[kk * HS + colbase];
      bf.y = hlds[(kk + 1) * HS + colbase];
#pragma unroll
      for (int mi = 0; mi < 4; mi++) {
        const float* wr = w6 + (size_t)((mg + mi) * 16 + ln) * 256 + kk;
        v2f af; af.x = wr[0]; af.y = wr[1];
        acc[mi] = __builtin_amdgcn_wmma_f32_16x16x4_f32(
            false, af, false, bf, (short)0, acc[mi], false, false);
      }
    }
#pragma unroll
    for (int mi = 0; mi < 4; mi++) {
#pragma unroll
      for (int i = 0; i < 8; i++) {
        const int m = (mg + mi) * 16 + i + 8 * hl;
        out[((size_t)b * 256 + m) * NPTS + n0 + colbase] = acc[mi][i] + b6[m];
      }
    }
  }
}

extern "C" void kernel_launch(void* const* d_in, const int* in_sizes, int n_in,
                              void* d_out, int out_size, void* d_ws, size_t ws_size,
                              hipStream_t stream) {
  const float* x   = (const float*)d_in[0];
  const float* w1  = (const float*)d_in[1];
  const float* b1  = (const float*)d_in[2];
  const float* w2  = (const float*)d_in[3];
  const float* b2  = (const float*)d_in[4];
  const float* w1a = (const float*)d_in[5];
  const float* b1a = (const float*)d_in[6];
  const float* w2a = (const float*)d_in[7];
  const float* b2a = (const float*)d_in[8];
  const float* w1b = (const float*)d_in[9];
  const float* b1b = (const float*)d_in[10];
  const float* w2b = (const float*)d_in[11];
  const float* b2b = (const float*)d_in[12];
  const float* w4  = (const float*)d_in[13];
  const float* b4  = (const float*)d_in[14];
  const float* w6  = (const float*)d_in[15];
  const float* b6  = (const float*)d_in[16];
  float* out = (float*)d_out;

  const int B = 16;
  char* ws = (char*)d_ws;
  int* idx2 = (int*)ws;                               // 16*4096 ints
  int* idx3 = (int*)(ws + (size_t)B * 4096 * 4);      // 16*2048 ints
  unsigned* ord0 = (unsigned*)(ws + (size_t)B * 4096 * 4 + (size_t)B * 2048 * 4);
  unsigned* ord1 = ord0 + B * 64;
  unsigned* ord2 = ord1 + B * 64;
  float* cbv = (float*)(ord2 + B * 64);               // 16*256 floats

  // 1) zero the ordered-max buffers (workspace is poisoned by harness)
  init_zero_kernel<<<(3 * B * 64 + 255) / 256, 256, 0, stream>>>(ord0, 3 * B * 64);

  // 2) furthest point sampling (branch 2: N/2, branch 3: N/4)
  fps_kernel<<<B, 1024, 0, stream>>>(x, idx2, 4096);
  fps_kernel<<<B, 1024, 0, stream>>>(x, idx3, 2048);

  // 3) branch convs -> per-batch gf max vectors
  branch_conv_kernel<<<dim3((8192 + 255) / 256, B), 256, 0, stream>>>(
      x, nullptr, 8192, w1, b1, w2, b2, ord0);
  branch_conv_kernel<<<dim3((4096 + 255) / 256, B), 256, 0, stream>>>(
      x, idx2, 4096, w1a, b1a, w2a, b2a, ord1);
  branch_conv_kernel<<<dim3((2048 + 255) / 256, B), 256, 0, stream>>>(
      x, idx3, 2048, w1b, b1b, w2b, b2b, ord2);

  // 4) fold gf vectors through w4[:,64:] + b4 into per-batch constant
  cb_kernel<<<B, 256, 0, stream>>>(ord0, ord1, ord2, w4, b4, cbv);

  // 5) fused heavy path with f32 WMMA
  main_gemm_kernel<<<dim3(8192 / 128, B), 256, 0, stream>>>(
      x, w1, b1, w2, b2, w4, cbv, w6, b6, out);
}